// MixedScoreMultiHeadAttention_62629213110573
// MI455X (gfx1250) — compile-verified
//
#include <hip/hip_runtime.h>
#include <hip/hip_bf16.h>

// MixedScoreMultiHeadAttention for MI455X (gfx1250), f32 WMMA path.
// B=4, R=C=256, EMBED=256, H=16, D=16, MS_HIDDEN=32.

#define EMB 256
#define NB 4
#define NH 16
#define HD 16
#define SEQ 256
#define MSH 32

typedef __attribute__((ext_vector_type(2))) float v2f;
typedef __attribute__((ext_vector_type(8))) float v8f;

// A/B fragment for V_WMMA_F32_16X16X4_F32:
// lanes 0-15 hold (k, k+1), lanes 16-31 hold (k+2, k+3) -> one contiguous
// float2 load at column offset k + 2*half.

// ---------------------------------------------------------------------------
// Kernel 1: Q = row_emb @ Wq^T, K = col_emb @ Wk^T, V = col_emb @ Wv^T
// One wave per 16x16 output tile; column tile == one head (HD==16).
// Output layout: [B][H][N][16] (head-major) for stage-2 locality.
// ---------------------------------------------------------------------------
__global__ __launch_bounds__(128) void qkv_proj_kernel(
    const float* __restrict__ row_emb, const float* __restrict__ col_emb,
    const float* __restrict__ Wq, const float* __restrict__ Wk,
    const float* __restrict__ Wv,
    float* __restrict__ Qws, float* __restrict__ Kws, float* __restrict__ Vws) {
  const int wave = (blockIdx.x * blockDim.x + threadIdx.x) >> 5;  // 0..3071
  const int lane = threadIdx.x & 31;
  const int n    = lane & 15;
  const int half = lane >> 4;

  // 1024 tiles per matrix: b(4) x rowTile(16) x head(16)
  const int which = wave >> 10;          // 0=Q, 1=K, 2=V
  const int rem   = wave & 1023;
  const int b     = rem >> 8;
  const int r2    = rem & 255;
  const int nTile = r2 >> 4;
  const int h     = r2 & 15;

  const float* src = (which == 0) ? row_emb : col_emb;
  const float* W   = (which == 0) ? Wq : ((which == 1) ? Wk : Wv);
  float*       dst = (which == 0) ? Qws : ((which == 1) ? Kws : Vws);

  const int n0 = nTile * 16;
  // A: src rows n0..n0+15 ; fragment column base = k + 2*half
  const float* srow = src + ((size_t)b * SEQ + (n0 + n)) * EMB + 2 * half;
  // B: B[k][col] = W[h*16+col][k]
  const float* wrow = W + (size_t)(h * HD + n) * EMB + 2 * half;

  v8f acc = {};
  #pragma unroll 8
  for (int k = 0; k < EMB; k += 4) {
    const float2 a2 = *reinterpret_cast<const float2*>(srow + k);
    const float2 w2 = *reinterpret_cast<const float2*>(wrow + k);
    v2f a, bb;
    a.x = a2.x;  a.y = a2.y;
    bb.x = w2.x; bb.y = w2.y;
    acc = __builtin_amdgcn_wmma_f32_16x16x4_f32(false, a, false, bb,
                                                (short)0, acc, false, false);
  }

  // C/D layout: VGPR v -> row = v + 8*half, col = lane&15
  float* drow = dst + (((size_t)b * NH + h) * SEQ) * HD;
  #pragma unroll
  for (int v = 0; v < 8; ++v) {
    const int r = n0 + v + half * 8;
    drow[(size_t)r * HD + n] = acc[v];
  }
}

// ---------------------------------------------------------------------------
// Kernel 2: per (b, h, 16-row block):
//   S = (Q Kt)*0.25 ; mixed = MLP(S, cost) ; P = softmax_row(mixed) ; O = P V
// One wave per block; scores staged through LDS to keep VGPRs low.
// ---------------------------------------------------------------------------
__global__ __launch_bounds__(32) void attn_mix_kernel(
    const float* __restrict__ Qws, const float* __restrict__ Kws,
    const float* __restrict__ Vws, const float* __restrict__ cost,
    const float* __restrict__ m1w, const float* __restrict__ m1b,
    const float* __restrict__ m2w, const float* __restrict__ m2b,
    float* __restrict__ out) {
  __shared__ float w1a[MSH], w1b[MSH], b1s[MSH], w2s[MSH];
  __shared__ float b2s;
  // Row stride 260: 260 % 64 == 4 -> 16-lane column reads hit 16 distinct
  // banks; rows stay 8B-aligned for float2 LDS reads.
  __shared__ float Wl[16][260];

  const int lane = threadIdx.x;
  const int n    = lane & 15;
  const int half = lane >> 4;

  const int bid   = blockIdx.x;     // B*H*16 = 1024
  const int b     = bid >> 8;
  const int rem   = bid & 255;
  const int h     = rem >> 4;
  const int rTile = rem & 15;
  const int r0    = rTile * 16;

  // Stage per-head MLP params into LDS (uniform broadcast reads later).
  w1a[lane] = m1w[(h * 2 + 0) * MSH + lane];
  w1b[lane] = m1w[(h * 2 + 1) * MSH + lane];
  b1s[lane] = m1b[h * MSH + lane];
  w2s[lane] = m2w[h * MSH + lane];
  if (lane == 0) b2s = m2b[h];
  __syncthreads();

  const float* Qh = Qws + (((size_t)b * NH + h) * SEQ) * HD;
  const float* Kh = Kws + (((size_t)b * NH + h) * SEQ) * HD;
  const float* Vh = Vws + (((size_t)b * NH + h) * SEQ) * HD;
  const float* ch = cost + ((size_t)b * SEQ) * SEQ;

  // Warm the cache for the V block (used only in the late output stage)
  // while we grind through the VALU-heavy MLP. -> global_prefetch_b8
  #pragma unroll
  for (int p = 0; p < 4; ++p)
    __builtin_prefetch(Vh + ((size_t)p * 32 + lane) * 32, 0, 0);

  // Preload Q-tile A fragments (16x16 -> four 16x4 chunks).
  v2f aq[4];
  #pragma unroll
  for (int j = 0; j < 4; ++j) {
    const float2 q2 = *reinterpret_cast<const float2*>(
        Qh + (size_t)(r0 + n) * HD + 4 * j + 2 * half);
    aq[j].x = q2.x;
    aq[j].y = q2.y;
  }

  float rmax[8];
  #pragma unroll
  for (int v = 0; v < 8; ++v) rmax[v] = -3.0e38f;
  const float bias2 = b2s;

  // ---- Pass 1: scores + mixed-score MLP -> LDS, tracking row max ----------
  for (int cc = 0; cc < 16; ++cc) {
    v8f acc = {};
    #pragma unroll
    for (int j = 0; j < 4; ++j) {
      const float2 k2 = *reinterpret_cast<const float2*>(
          Kh + (size_t)(cc * 16 + n) * HD + 4 * j + 2 * half);
      v2f bb;
      bb.x = k2.x;
      bb.y = k2.y;
      acc = __builtin_amdgcn_wmma_f32_16x16x4_f32(false, aq[j], false, bb,
                                                  (short)0, acc, false, false);
    }
    // hidden = relu(dot*w1a + cost*w1b + b1); mixed = hidden.w2 + b2
    float dv[8], cv[8], hs[8];
    #pragma unroll
    for (int v = 0; v < 8; ++v) {
      const int r = r0 + v + half * 8;
      dv[v] = acc[v] * 0.25f;  // 1/sqrt(QKV_DIM=16)
      cv[v] = ch[(size_t)r * SEQ + cc * 16 + n];
      hs[v] = bias2;
    }
    for (int m = 0; m < MSH; ++m) {
      const float A1 = w1a[m], B1 = w1b[m], C1 = b1s[m], W2 = w2s[m];
      #pragma unroll
      for (int v = 0; v < 8; ++v) {
        float t = fmaf(dv[v], A1, fmaf(cv[v], B1, C1));
        t = fmaxf(t, 0.0f);
        hs[v] = fmaf(t, W2, hs[v]);
      }
    }
    #pragma unroll
    for (int v = 0; v < 8; ++v) {
      Wl[v + half * 8][cc * 16 + n] = hs[v];
      rmax[v] = fmaxf(rmax[v], hs[v]);
    }
  }

  // Rows live per (v, half); reduce across the 16-lane half.
  #pragma unroll
  for (int v = 0; v < 8; ++v) {
    rmax[v] = fmaxf(rmax[v], __shfl_xor(rmax[v], 1, 32));
    rmax[v] = fmaxf(rmax[v], __shfl_xor(rmax[v], 2, 32));
    rmax[v] = fmaxf(rmax[v], __shfl_xor(rmax[v], 4, 32));
    rmax[v] = fmaxf(rmax[v], __shfl_xor(rmax[v], 8, 32));
  }

  // ---- Pass 2: exp in place, accumulate row sums --------------------------
  float rsum[8];
  #pragma unroll
  for (int v = 0; v < 8; ++v) rsum[v] = 0.0f;
  for (int cc = 0; cc < 16; ++cc) {
    #pragma unroll
    for (int v = 0; v < 8; ++v) {
      const float e = __expf(Wl[v + half * 8][cc * 16 + n] - rmax[v]);
      Wl[v + half * 8][cc * 16 + n] = e;
      rsum[v] += e;
    }
  }
  #pragma unroll
  for (int v = 0; v < 8; ++v) {
    rsum[v] += __shfl_xor(rsum[v], 1, 32);
    rsum[v] += __shfl_xor(rsum[v], 2, 32);
    rsum[v] += __shfl_xor(rsum[v], 4, 32);
    rsum[v] += __shfl_xor(rsum[v], 8, 32);
    rsum[v] = 1.0f / rsum[v];  // folded into oacc after the PV WMMAs
  }
  __syncthreads();

  // ---- O = exp(S') @ V, then scale rows by 1/rsum -------------------------
  v8f oacc = {};
  for (int c0 = 0; c0 < SEQ; c0 += 4) {
    const int k0 = c0 + 2 * half;
    const float2 w2 = *reinterpret_cast<const float2*>(&Wl[n][k0]);
    v2f aw, bv;
    aw.x = w2.x;
    aw.y = w2.y;
    bv.x = Vh[(size_t)k0 * HD + n];
    bv.y = Vh[(size_t)(k0 + 1) * HD + n];
    oacc = __builtin_amdgcn_wmma_f32_16x16x4_f32(false, aw, false, bv,
                                                 (short)0, oacc, false, false);
  }

  // out[b][r][h*16 + d], rows of oacc match rsum's (v, half) mapping.
  #pragma unroll
  for (int v = 0; v < 8; ++v) {
    const int r = r0 + v + half * 8;
    out[((size_t)b * SEQ + r) * EMB + h * HD + n] = oacc[v] * rsum[v];
  }
}

extern "C" void kernel_launch(void* const* d_in, const int* in_sizes, int n_in,
                              void* d_out, int out_size, void* d_ws,
                              size_t ws_size, hipStream_t stream) {
  const float* row_emb = (const float*)d_in[0];
  const float* col_emb = (const float*)d_in[1];
  const float* cost    = (const float*)d_in[2];
  const float* Wq      = (const float*)d_in[3];
  const float* Wk      = (const float*)d_in[4];
  const float* Wv      = (const float*)d_in[5];
  const float* m1w     = (const float*)d_in[6];
  const float* m1b     = (const float*)d_in[7];
  const float* m2w     = (const float*)d_in[8];
  const float* m2b     = (const float*)d_in[9];
  float* out = (float*)d_out;

  const size_t perMat = (size_t)NB * NH * SEQ * HD;  // 262144 floats
  float* Qws = (float*)d_ws;
  float* Kws = Qws + perMat;
  float* Vws = Kws + perMat;

  // 3 matrices x 1024 tiles = 3072 waves; 4 waves per block.
  qkv_proj_kernel<<<768, 128, 0, stream>>>(row_emb, col_emb, Wq, Wk, Wv, Qws,
                                           Kws, Vws);
  // B*H*(R/16) = 1024 wave-blocks.
  attn_mix_kernel<<<1024, 32, 0, stream>>>(Qws, Kws, Vws, cost, m1w, m1b, m2w,
                                           m2b, out);
}